// SpikingLayer_90202903151092
// MI455X (gfx1250) — compile-verified
//
#include <hip/hip_runtime.h>

typedef float v2f __attribute__((ext_vector_type(2)));
typedef float v8f __attribute__((ext_vector_type(8)));

#define T_STEPS 500
#define N_ROWS  65536          // B*N = 16*4096
#define KLEN    149            // EPSP kernel taps
#define NCHUNK  41             // K window 164 = 41 * 4
#define NTILE   32             // ceil(500/16)
#define ALPHA   0.90483741803595952f   // exp(-1/10)

__launch_bounds__(256)
__global__ void snn_wmma_kernel(const float* __restrict__ x,
                                const float* __restrict__ kern,
                                float* __restrict__ out)
{
    __shared__ float lds[8 * 256];          // one 16x16 f32 tile per wave
    const int lane  = threadIdx.x & 31;
    const int ws    = threadIdx.x >> 5;
    const int waveG = blockIdx.x * 8 + ws;
    const int r0    = waveG * 16;           // first of this wave's 16 rows

    const int h   = lane >> 4;              // lane half (K split for A/B frags)
    const int j   = lane & 15;              // M index for A, N index for B/C
    const int row = r0 + j;                 // this lane's input row

    // ---- Constant Toeplitz B fragments: Bm[p][j] = k[148+j-p] ----
    // chunk c, VGPR0 holds K = 2h (p = 4c+2h), VGPR1 holds K = 2h+1.
    float b0[NCHUNK], b1[NCHUNK];
#pragma unroll
    for (int c = 0; c < NCHUNK; ++c) {
        int i0 = 148 + j - (4 * c + 2 * h);
        int i1 = i0 - 1;
        b0[c] = (i0 >= 0 && i0 < KLEN) ? kern[i0] : 0.0f;
        b1[c] = (i1 >= 0 && i1 < KLEN) ? kern[i1] : 0.0f;
    }

    // ---- Rolling A window for tile 0: chunk c covers u = -148 + 4c + kk ----
    // c < 37 -> u < 0 -> zero padding; c = 37..40 -> u = 0..15 (all valid).
    float aA[NCHUNK], aB[NCHUNK];
#pragma unroll
    for (int c = 0; c < NCHUNK - 4; ++c) { aA[c] = 0.0f; aB[c] = 0.0f; }
#pragma unroll
    for (int c = NCHUNK - 4; c < NCHUNK; ++c) {
        unsigned u = (unsigned)(4 * (c - (NCHUNK - 4)) + 2 * h);
        aA[c] = x[u * N_ROWS + row];
        aB[c] = x[(u + 1) * N_ROWS + row];
    }

    float  rstate = 0.0f;                   // refractory state (lane = row)
    float* wtile  = &lds[ws * 256];

#pragma unroll 1
    for (int tb = 0; tb < NTILE; ++tb) {
        const int t0 = tb * 16;

        // ---- vmem tile: D[16 rows x 16 times], two interleaved WMMA chains ----
        v8f acc0 = {}, acc1 = {};
#pragma unroll
        for (int c = 0; c < NCHUNK; c += 2) {
            v2f a; a.x = aA[c]; a.y = aB[c];
            v2f b; b.x = b0[c]; b.y = b1[c];
            acc0 = __builtin_amdgcn_wmma_f32_16x16x4_f32(
                false, a, false, b, (short)0, acc0, false, false);
            if (c + 1 < NCHUNK) {
                v2f a2; a2.x = aA[c + 1]; a2.y = aB[c + 1];
                v2f b2; b2.x = b0[c + 1]; b2.y = b1[c + 1];
                acc1 = __builtin_amdgcn_wmma_f32_16x16x4_f32(
                    false, a2, false, b2, (short)0, acc1, false, false);
            }
        }
        v8f acc = acc0 + acc1;

        // ---- Shift window by 4 chunks and issue the 16 fresh-time loads now,
        //      so they overlap the LDS transpose + scan below. Branchless
        //      clamp+select handles the u >= 500 tail (no scalar branches). ----
        if (tb < NTILE - 1) {
#pragma unroll
            for (int c = 0; c < NCHUNK - 4; ++c) { aA[c] = aA[c + 4]; aB[c] = aB[c + 4]; }
#pragma unroll
            for (int c = NCHUNK - 4; c < NCHUNK; ++c) {
                unsigned u0 = (unsigned)(t0 + 16 + 4 * (c - (NCHUNK - 4)) + 2 * h);
                unsigned u1 = u0 + 1;
                unsigned c0 = (u0 < T_STEPS) ? u0 : (T_STEPS - 1);
                unsigned c1 = (u1 < T_STEPS) ? u1 : (T_STEPS - 1);
                float v0 = x[c0 * N_ROWS + row];
                float v1 = x[c1 * N_ROWS + row];
                aA[c] = (u0 < T_STEPS) ? v0 : 0.0f;
                aB[c] = (u1 < T_STEPS) ? v1 : 0.0f;
            }
        }

        // ---- Transpose C tile through wave-private LDS ----
        // acc[v] holds (M = v + 8*h, N = j)
#pragma unroll
        for (int v = 0; v < 8; ++v)
            wtile[(v + 8 * h) * 16 + j] = acc[v];
        asm volatile("s_wait_dscnt 0" ::: "memory");

        // ---- Sequential threshold + refractory scan ----
        // Lane L (0..15) owns row r0+j; upper half mirrors (never stores).
#pragma unroll
        for (int jj = 0; jj < 16; ++jj) {
            float v = wtile[j * 16 + jj];
            float s = (v + rstate >= 1.0f) ? 1.0f : 0.0f;
            rstate  = (rstate - s) * ALPHA;
            int t = t0 + jj;
            if (h == 0 && t < T_STEPS)
                out[(size_t)t * N_ROWS + row] = s;      // coalesced 64B row
        }
        asm volatile("s_wait_dscnt 0" ::: "memory");
    }
}

extern "C" void kernel_launch(void* const* d_in, const int* in_sizes, int n_in,
                              void* d_out, int out_size, void* d_ws, size_t ws_size,
                              hipStream_t stream) {
    const float* x    = (const float*)d_in[0];   // (T, B, 1, N) spikes
    const float* kern = (const float*)d_in[1];   // (1, 149) EPSP kernel
    float*       out  = (float*)d_out;           // (T, B, N) spikes

    (void)in_sizes; (void)n_in; (void)out_size; (void)d_ws; (void)ws_size;

    // 65536 rows / 16 rows-per-wave / 8 waves-per-block = 512 blocks
    snn_wmma_kernel<<<dim3(N_ROWS / 16 / 8), dim3(256), 0, stream>>>(x, kern, out);
}